// GraphAttentionLayer_87720412053518
// MI455X (gfx1250) — compile-verified
//
#include <hip/hip_runtime.h>
#include <math.h>

#define N_NODES 100000
#define E_EDGES 3200000
#define IN_FEAT 128
#define OUT_FEAT 64
#define ALPHA 0.2f
#define LN_EPS 1e-5f

typedef __attribute__((ext_vector_type(2))) float v2f;
typedef __attribute__((ext_vector_type(8))) float v8f;

// ---------------------------------------------------------------------------
// Kernel 0: zero h_prime accumulator (must be re-zeroed every call).
// ---------------------------------------------------------------------------
__global__ void gat_zero_kernel(float4* __restrict__ p, int n4) {
    int i = blockIdx.x * blockDim.x + threadIdx.x;
    if (i < n4) p[i] = make_float4(0.f, 0.f, 0.f, 0.f);
}

// ---------------------------------------------------------------------------
// Kernel 1: h = x @ W^T + b via V_WMMA_F32_16X16X4_F32, fused with
//           sl = h @ a[:64], sr = h @ a[64:]  (LDS ds_add_f32 reduction).
// Block = 128 threads (4 waves). Block owns a 16-row node tile; wave w owns
// output columns [16w, 16w+16). K = 128 swept in steps of 4.
//
// fp32 WMMA VGPR layouts (ISA 7.12.2):
//   A 16x4 : lane<16 -> M=lane,   v0=K0,v1=K1 ; lane>=16 -> M=lane-16, v0=K2,v1=K3
//   B 4x16 : lane<16 -> N=lane,   v0=K0,v1=K1 ; lane>=16 -> N=lane-16, v0=K2,v1=K3
//   C 16x16: vgpr r, lane l -> M = r + 8*(l>=16), N = l&15
// ---------------------------------------------------------------------------
__global__ __launch_bounds__(128) void gat_linear_kernel(
    const float* __restrict__ x, const float* __restrict__ W,
    const float* __restrict__ bias, const float* __restrict__ a,
    float* __restrict__ h, float* __restrict__ sl, float* __restrict__ sr)
{
    __shared__ float slds[16];
    __shared__ float srlds[16];

    const int tid  = threadIdx.x;
    const int lane = tid & 31;
    const int wave = tid >> 5;          // 0..3
    const int row0 = blockIdx.x * 16;   // 6250 * 16 == 100000 exactly
    const int col0 = wave * 16;

    if (tid < 16) { slds[tid] = 0.f; srlds[tid] = 0.f; }
    __syncthreads();

    const int l15  = lane & 15;
    const int half = lane >> 4;

    // per-lane bases: float2 (K, K+1) loads, halves offset by +2 in K
    const float* aptr = x + (size_t)(row0 + l15) * IN_FEAT + half * 2;
    const float* bptr = W + (size_t)(col0 + l15) * IN_FEAT + half * 2;

    v8f c = {};
#pragma unroll
    for (int k0 = 0; k0 < IN_FEAT; k0 += 4) {
        v2f av = *(const v2f*)(aptr + k0);
        v2f bv = *(const v2f*)(bptr + k0);
        // 8-arg form: (neg_a, A, neg_b, B, c_mod, C, reuse_a, reuse_b)
        c = __builtin_amdgcn_wmma_f32_16x16x4_f32(
                false, av, false, bv, (short)0, c, false, false);
    }

    const int   col = col0 + l15;
    const float bv  = bias[col];
    const float aL  = a[col];
    const float aR  = a[OUT_FEAT + col];

#pragma unroll
    for (int r = 0; r < 8; ++r) {
        const int   row = r + 8 * half;
        const float v   = c[r] + bv;
        h[(size_t)(row0 + row) * OUT_FEAT + col] = v;
        atomicAdd(&slds[row], v * aL);   // ds_add_f32
        atomicAdd(&srlds[row], v * aR);  // ds_add_f32
    }
    __syncthreads();

    if (tid < 16) {
        sl[row0 + tid] = slds[tid];
        sr[row0 + tid] = srlds[tid];
    }
}

// ---------------------------------------------------------------------------
// Kernel 2: per-edge attention weight + scatter SpMM.
// Wave-per-edge: 32 lanes x float2 = 64 features. h and h_prime are
// L2-resident (25.6 MB each << 192 MB L2), scatter via no-return
// global_atomic_add_f32.
// ---------------------------------------------------------------------------
__global__ __launch_bounds__(256) void gat_edge_kernel(
    const int* __restrict__ src, const int* __restrict__ dst,
    const float* __restrict__ sl, const float* __restrict__ sr,
    const float* __restrict__ h, float* __restrict__ hp)
{
    const int lane   = threadIdx.x & 31;
    const int wave   = (blockIdx.x * blockDim.x + threadIdx.x) >> 5;
    const int nwaves = (gridDim.x * blockDim.x) >> 5;

    for (int e = wave; e < E_EDGES; e += nwaves) {
        const int s = src[e];
        const int d = dst[e];

        const float score = sl[s] + sr[d];
        const float lr    = score > 0.f ? score : ALPHA * score;
        const float w     = __expf(-lr);

        const float2 v = ((const float2*)(h + (size_t)d * OUT_FEAT))[lane];
        float* o = hp + (size_t)s * OUT_FEAT + 2 * lane;
        unsafeAtomicAdd(o + 0, w * v.x);   // global_atomic_add_f32 (no return)
        unsafeAtomicAdd(o + 1, w * v.y);
    }
}

// ---------------------------------------------------------------------------
// Kernel 3: LayerNorm (biased var) + ELU, wave-per-node (wave32).
// ---------------------------------------------------------------------------
__global__ __launch_bounds__(256) void gat_norm_kernel(
    const float* __restrict__ hp, const float* __restrict__ gamma,
    const float* __restrict__ beta, float* __restrict__ out)
{
    const int lane = threadIdx.x & 31;
    const int node = (blockIdx.x * blockDim.x + threadIdx.x) >> 5;
    if (node >= N_NODES) return;

    const float2 v = ((const float2*)(hp + (size_t)node * OUT_FEAT))[lane];
    float s  = v.x + v.y;
    float ss = v.x * v.x + v.y * v.y;
#pragma unroll
    for (int m = 16; m >= 1; m >>= 1) {
        s  += __shfl_xor(s, m, 32);
        ss += __shfl_xor(ss, m, 32);
    }
    const float mu  = s * (1.f / OUT_FEAT);
    const float var = ss * (1.f / OUT_FEAT) - mu * mu;
    const float rs  = rsqrtf(var + LN_EPS);

    float y0 = (v.x - mu) * rs * gamma[2 * lane + 0] + beta[2 * lane + 0];
    float y1 = (v.y - mu) * rs * gamma[2 * lane + 1] + beta[2 * lane + 1];
    y0 = y0 > 0.f ? y0 : __expf(y0) - 1.f;   // ELU
    y1 = y1 > 0.f ? y1 : __expf(y1) - 1.f;

    float2 r; r.x = y0; r.y = y1;
    ((float2*)(out + (size_t)node * OUT_FEAT))[lane] = r;
}

// ---------------------------------------------------------------------------
// Launch
// ---------------------------------------------------------------------------
extern "C" void kernel_launch(void* const* d_in, const int* in_sizes, int n_in,
                              void* d_out, int out_size, void* d_ws, size_t ws_size,
                              hipStream_t stream) {
    (void)in_sizes; (void)n_in; (void)out_size; (void)ws_size;

    const float* x     = (const float*)d_in[0];
    const int*   edge  = (const int*)  d_in[1];   // [2, E]
    const float* W     = (const float*)d_in[2];
    const float* b     = (const float*)d_in[3];
    const float* a     = (const float*)d_in[4];
    const float* gamma = (const float*)d_in[5];
    const float* beta  = (const float*)d_in[6];
    float*       out   = (float*)d_out;

    float* ws = (float*)d_ws;
    float* h  = ws;                                  // N*64
    float* hp = ws + (size_t)N_NODES * OUT_FEAT;     // N*64
    float* sl = ws + 2 * (size_t)N_NODES * OUT_FEAT; // N
    float* sr = sl + N_NODES;                        // N

    const int n4 = (N_NODES * OUT_FEAT) / 4;
    gat_zero_kernel<<<(n4 + 255) / 256, 256, 0, stream>>>((float4*)hp, n4);

    gat_linear_kernel<<<N_NODES / 16, 128, 0, stream>>>(x, W, b, a, h, sl, sr);

    gat_edge_kernel<<<4096, 256, 0, stream>>>(edge, edge + E_EDGES, sl, sr, h, hp);

    gat_norm_kernel<<<(N_NODES * 32 + 255) / 256, 256, 0, stream>>>(hp, gamma, beta, out);
}